// Model_68856915690095
// MI455X (gfx1250) — compile-verified
//
#include <hip/hip_runtime.h>

typedef float v2f __attribute__((ext_vector_type(2)));
typedef float v8f __attribute__((ext_vector_type(8)));

#define N_T 200000
#define N_M 20000
#define DD 128
#define D_IN 384
#define E_EDGES 1000000
#define EL_EDGES 500000

// ================= CSR construction (per call; edges are layer-invariant) =================

__global__ void degree_kernel(const int* __restrict__ src, const int* __restrict__ dst,
                              int* __restrict__ degT, int* __restrict__ degM, int E) {
    int e = blockIdx.x * blockDim.x + threadIdx.x;
    if (e < E) {
        atomicAdd(&degT[src[e]], 1);
        atomicAdd(&degM[dst[e]], 1);
    }
}

// per-block sum of a 256-chunk of deg
__global__ void reduce_kernel(const int* __restrict__ deg, int n, int* __restrict__ bsum) {
    __shared__ int s[256];
    int tid = threadIdx.x;
    int i = blockIdx.x * 256 + tid;
    s[tid] = (i < n) ? deg[i] : 0;
    __syncthreads();
    for (int off = 128; off > 0; off >>= 1) {
        if (tid < off) s[tid] += s[tid + off];
        __syncthreads();
    }
    if (tid == 0) bsum[blockIdx.x] = s[0];
}

// single-block exclusive scan of block sums (nB <= 1024)
__global__ __launch_bounds__(1024) void scan_small_kernel(const int* __restrict__ in, int nB,
                                                          int* __restrict__ out) {
    __shared__ int s[1024];
    int tid = threadIdx.x;
    int v = (tid < nB) ? in[tid] : 0;
    s[tid] = v;
    __syncthreads();
    for (int off = 1; off < 1024; off <<= 1) {
        int t2 = (tid >= off) ? s[tid - off] : 0;
        __syncthreads();
        s[tid] += t2;
        __syncthreads();
    }
    if (tid < nB) out[tid] = s[tid] - v;   // exclusive
}

// per-block exclusive scan + block offset -> rowStart
__global__ void scan_apply_kernel(const int* __restrict__ deg, int n,
                                  const int* __restrict__ boff, int* __restrict__ rowStart) {
    __shared__ int s[256];
    int tid = threadIdx.x;
    int i = blockIdx.x * 256 + tid;
    int v = (i < n) ? deg[i] : 0;
    s[tid] = v;
    __syncthreads();
    for (int off = 1; off < 256; off <<= 1) {
        int t2 = (tid >= off) ? s[tid - off] : 0;
        __syncthreads();
        s[tid] += t2;
        __syncthreads();
    }
    if (i < n) rowStart[i] = s[tid] - v + boff[blockIdx.x];
}

// bucket fill: T-buckets hold mentor ids grouped by thesis; M-buckets hold thesis ids grouped by mentor
__global__ void fill_kernel(const int* __restrict__ src, const int* __restrict__ dst,
                            const int* __restrict__ rsT, const int* __restrict__ rsM,
                            int* __restrict__ curT, int* __restrict__ curM,
                            int* __restrict__ bktT, int* __restrict__ bktM, int E) {
    int e = blockIdx.x * blockDim.x + threadIdx.x;
    if (e >= E) return;
    int s = src[e];
    int d = dst[e];
    int pm = atomicAdd(&curM[d], 1);
    bktM[rsM[d] + pm] = s;
    int pt = atomicAdd(&curT[s], 1);
    bktT[rsT[s] + pt] = d;
}

// ================= gather-mean: agg[node] = mean(feat[bucket[node]]) =================
// wave per node, float4 per lane, 2-way unrolled neighbor loop (pure streaming reads)
__global__ void gather_mean_kernel(const float* __restrict__ feat, const int* __restrict__ bucket,
                                   const int* __restrict__ rowStart, const int* __restrict__ deg,
                                   float* __restrict__ agg, int n) {
    int t = blockIdx.x * blockDim.x + threadIdx.x;
    int node = t >> 5;
    int lane = t & 31;
    if (node >= n) return;
    int start = rowStart[node];
    int d = deg[node];
    float ax = 0.f, ay = 0.f, az = 0.f, aw = 0.f;
    int j = 0;
    for (; j + 1 < d; j += 2) {
        int nb0 = bucket[start + j];
        int nb1 = bucket[start + j + 1];
        float4 v0 = *(const float4*)(feat + (size_t)nb0 * DD + lane * 4);
        float4 v1 = *(const float4*)(feat + (size_t)nb1 * DD + lane * 4);
        ax += v0.x + v1.x;
        ay += v0.y + v1.y;
        az += v0.z + v1.z;
        aw += v0.w + v1.w;
    }
    if (j < d) {
        int nb = bucket[start + j];
        float4 v = *(const float4*)(feat + (size_t)nb * DD + lane * 4);
        ax += v.x; ay += v.y; az += v.z; aw += v.w;
    }
    float invd = (d > 0) ? (1.0f / (float)d) : 0.0f;   // isolated node -> 0, as reference
    float4 o;
    o.x = ax * invd; o.y = ay * invd; o.z = az * invd; o.w = aw * invd;
    *(float4*)(agg + (size_t)node * DD + lane * 4) = o;
}

// ================= x_t = thesis_x @ lin_W + lin_b + thesis_emb (fp32 WMMA) =================
// block = 8 waves; block owns 16 rows, wave w owns cols [16w,16w+16)
// A tile (16x384) staged in LDS, padded stride 388 (4*r mod 64 banks -> conflict-free)
__global__ __launch_bounds__(256) void lin_kernel(const float* __restrict__ X,
                                                  const float* __restrict__ W,
                                                  const float* __restrict__ bvec,
                                                  const float* __restrict__ emb,
                                                  float* __restrict__ out) {
    __shared__ float As[16 * 388];
    const int rowBase = blockIdx.x * 16;
    for (int i = threadIdx.x; i < 16 * 96; i += 256) {            // 16x384 as float4
        int r = i / 96, k4 = i % 96;
        float4 v = *(const float4*)(X + (size_t)(rowBase + r) * D_IN + k4 * 4);
        *(float4*)&As[r * 388 + k4 * 4] = v;
    }
    __syncthreads();

    const int lane = threadIdx.x & 31;
    const int w = threadIdx.x >> 5;
    const int colBase = w * 16;
    const int n = lane & 15, hi = lane >> 4;

    v8f acc = {0.f, 0.f, 0.f, 0.f, 0.f, 0.f, 0.f, 0.f};
    #pragma unroll 4
    for (int k0 = 0; k0 < D_IN; k0 += 4) {
        const int kk = k0 + 2 * hi;
        float2 a2 = *(const float2*)&As[n * 388 + kk];
        v2f a; a[0] = a2.x; a[1] = a2.y;
        const float* Bp = W + (size_t)kk * DD + colBase + n;
        v2f bb; bb[0] = Bp[0]; bb[1] = Bp[DD];
        acc = __builtin_amdgcn_wmma_f32_16x16x4_f32(false, a, false, bb, (short)0, acc, false, false);
    }
    const int col = colBase + n;
    const float bias = bvec[col];
    #pragma unroll
    for (int r = 0; r < 8; ++r) {
        int row = rowBase + r + 8 * hi;
        out[(size_t)row * DD + col] = acc[r] + bias + emb[(size_t)row * DD + col];
    }
}

// ================= out = agg @ Wl + x @ Wr + b (+ReLU) (fp32 WMMA, fused) =================
__global__ __launch_bounds__(256) void sage_gemm_kernel(const float* __restrict__ agg,
                                                        const float* __restrict__ Wl,
                                                        const float* __restrict__ bvec,
                                                        const float* __restrict__ x,
                                                        const float* __restrict__ Wr,
                                                        float* __restrict__ out,
                                                        int relu) {
    __shared__ float A1[16 * 132];   // mean-neighbor features
    __shared__ float A2[16 * 132];   // self features
    const int rowBase = blockIdx.x * 16;
    for (int i = threadIdx.x; i < 16 * 32; i += 256) {            // 16x128 as float4
        int r = i / 32, k4 = i % 32;
        *(float4*)&A1[r * 132 + k4 * 4] = *(const float4*)(agg + (size_t)(rowBase + r) * DD + k4 * 4);
        *(float4*)&A2[r * 132 + k4 * 4] = *(const float4*)(x + (size_t)(rowBase + r) * DD + k4 * 4);
    }
    __syncthreads();

    const int lane = threadIdx.x & 31;
    const int w = threadIdx.x >> 5;
    const int colBase = w * 16;
    const int n = lane & 15, hi = lane >> 4;

    v8f acc = {0.f, 0.f, 0.f, 0.f, 0.f, 0.f, 0.f, 0.f};
    #pragma unroll 4
    for (int k0 = 0; k0 < DD; k0 += 4) {
        const int kk = k0 + 2 * hi;
        {
            float2 a2 = *(const float2*)&A1[n * 132 + kk];
            v2f a; a[0] = a2.x; a[1] = a2.y;
            const float* Bp = Wl + (size_t)kk * DD + colBase + n;
            v2f bb; bb[0] = Bp[0]; bb[1] = Bp[DD];
            acc = __builtin_amdgcn_wmma_f32_16x16x4_f32(false, a, false, bb, (short)0, acc, false, false);
        }
        {
            float2 a2 = *(const float2*)&A2[n * 132 + kk];
            v2f a; a[0] = a2.x; a[1] = a2.y;
            const float* Bp = Wr + (size_t)kk * DD + colBase + n;
            v2f bb; bb[0] = Bp[0]; bb[1] = Bp[DD];
            acc = __builtin_amdgcn_wmma_f32_16x16x4_f32(false, a, false, bb, (short)0, acc, false, false);
        }
    }
    const int col = colBase + n;
    const float bias = bvec[col];
    #pragma unroll
    for (int r = 0; r < 8; ++r) {
        int row = rowBase + r + 8 * hi;
        float v = acc[r] + bias;
        if (relu) v = fmaxf(v, 0.0f);
        out[(size_t)row * DD + col] = v;
    }
}

// ================= out[e] = dot(x_t[ls[e]], x_m[ld[e]]) =================
__global__ void dot_kernel(const float* __restrict__ xt, const float* __restrict__ xm,
                           const int* __restrict__ ls, const int* __restrict__ ld,
                           float* __restrict__ out, int n) {
    int t = blockIdx.x * blockDim.x + threadIdx.x;
    int e = t >> 5;
    int lane = t & 31;
    if (e >= n) return;
    int s = ls[e];
    int d = ld[e];
    float4 a = *(const float4*)(xt + (size_t)s * DD + lane * 4);
    float4 b = *(const float4*)(xm + (size_t)d * DD + lane * 4);
    float v = a.x * b.x + a.y * b.y + a.z * b.z + a.w * b.w;
    #pragma unroll
    for (int off = 16; off > 0; off >>= 1) v += __shfl_xor(v, off, 32);
    if (lane == 0) out[e] = v;
}

extern "C" void kernel_launch(void* const* d_in, const int* in_sizes, int n_in,
                              void* d_out, int out_size, void* d_ws, size_t ws_size,
                              hipStream_t stream) {
    const float* thesis_x   = (const float*)d_in[0];
    const float* lin_W      = (const float*)d_in[1];
    const float* lin_b      = (const float*)d_in[2];
    const float* thesis_emb = (const float*)d_in[3];
    const float* mentor_emb = (const float*)d_in[4];
    const float* W_l_t2m    = (const float*)d_in[5];
    const float* b_l_t2m    = (const float*)d_in[6];
    const float* W_r_t2m    = (const float*)d_in[7];
    const float* W_l_m2t    = (const float*)d_in[8];
    const float* b_l_m2t    = (const float*)d_in[9];
    const float* W_r_m2t    = (const float*)d_in[10];
    const int* edge_src     = (const int*)d_in[11];
    const int* edge_dst     = (const int*)d_in[12];
    const int* label_src    = (const int*)d_in[13];
    const int* label_dst    = (const int*)d_in[14];
    float* out = (float*)d_out;

    const size_t TN = (size_t)N_T * DD;   // 25,600,000 floats
    const size_t MN = (size_t)N_M * DD;   //  2,560,000 floats

    float* fws  = (float*)d_ws;
    float* t0   = fws;
    float* t1   = t0 + TN;
    float* aggT = t1 + TN;
    float* m0   = aggT + TN;
    float* m1   = m0 + MN;
    float* aggM = m1 + MN;
    int* ip     = (int*)(aggM + MN);
    int* degT   = ip;               // N_T
    int* degM   = degT + N_T;       // N_M (contiguous with degT for one memset)
    int* rsT    = degM + N_M;       // N_T
    int* rsM    = rsT + N_T;        // N_M
    int* curT   = rsM + N_M;        // N_T
    int* curM   = curT + N_T;       // N_M (contiguous with curT for one memset)
    int* bktT   = curM + N_M;       // E
    int* bktM   = bktT + E_EDGES;   // E
    int* bsum   = bktM + E_EDGES;   // <=1024
    int* boff   = bsum + 1024;      // <=1024

    const int eBlocks = (E_EDGES + 255) / 256;     // 3907
    const int tBlocks = (N_T + 255) / 256;         // 782  (<=1024 for small scan)
    const int mBlocks = (N_M + 255) / 256;         // 79   (<=1024 for small scan)

    // ---- CSR build ----
    hipMemsetAsync(degT, 0, (size_t)(N_T + N_M) * sizeof(int), stream);
    degree_kernel<<<eBlocks, 256, 0, stream>>>(edge_src, edge_dst, degT, degM, E_EDGES);

    reduce_kernel<<<tBlocks, 256, 0, stream>>>(degT, N_T, bsum);
    scan_small_kernel<<<1, 1024, 0, stream>>>(bsum, tBlocks, boff);
    scan_apply_kernel<<<tBlocks, 256, 0, stream>>>(degT, N_T, boff, rsT);

    reduce_kernel<<<mBlocks, 256, 0, stream>>>(degM, N_M, bsum);
    scan_small_kernel<<<1, 1024, 0, stream>>>(bsum, mBlocks, boff);
    scan_apply_kernel<<<mBlocks, 256, 0, stream>>>(degM, N_M, boff, rsM);

    hipMemsetAsync(curT, 0, (size_t)(N_T + N_M) * sizeof(int), stream);
    fill_kernel<<<eBlocks, 256, 0, stream>>>(edge_src, edge_dst, rsT, rsM, curT, curM,
                                             bktT, bktM, E_EDGES);

    // ---- input projection: x_t = thesis_x @ lin_W + lin_b + thesis_emb ; x_m = mentor_emb ----
    lin_kernel<<<N_T / 16, 256, 0, stream>>>(thesis_x, lin_W, lin_b, thesis_emb, t0);
    hipMemcpyAsync(m0, mentor_emb, MN * sizeof(float), hipMemcpyDeviceToDevice, stream);

    // ---- 2 SAGE layers (ping-pong) ----
    for (int l = 0; l < 2; ++l) {
        float* xt = l ? t1 : t0;
        float* xm = l ? m1 : m0;
        float* nt = l ? t0 : t1;
        float* nm = l ? m0 : m1;
        gather_mean_kernel<<<(N_M * 32) / 256, 256, 0, stream>>>(xt, bktM, rsM, degM, aggM, N_M);
        gather_mean_kernel<<<(N_T * 32) / 256, 256, 0, stream>>>(xm, bktT, rsT, degT, aggT, N_T);
        sage_gemm_kernel<<<N_M / 16, 256, 0, stream>>>(aggM,
                                                       W_l_t2m + (size_t)l * DD * DD,
                                                       b_l_t2m + (size_t)l * DD,
                                                       xm,
                                                       W_r_t2m + (size_t)l * DD * DD,
                                                       nm, l == 0 ? 1 : 0);
        sage_gemm_kernel<<<N_T / 16, 256, 0, stream>>>(aggT,
                                                       W_l_m2t + (size_t)l * DD * DD,
                                                       b_l_m2t + (size_t)l * DD,
                                                       xt,
                                                       W_r_m2t + (size_t)l * DD * DD,
                                                       nt, l == 0 ? 1 : 0);
    }

    // after l=1 results are back in t0 / m0
    dot_kernel<<<(EL_EDGES * 32) / 256, 256, 0, stream>>>(t0, m0, label_src, label_dst, out, EL_EDGES);
}